// unit_gcn_23613730193751
// MI455X (gfx1250) — compile-verified
//
#include <hip/hip_runtime.h>

// ---------------------------------------------------------------------------
// unit_gcn for MI455X (gfx1250, wave32, WMMA).
//
// y[n,o,t,w] = sum_k sum_c W[k,o,c] * ( sum_v x[n,c,t,v] * A[k,v,w] )
// then training-mode BatchNorm over (N,T,V) + ReLU.
//
// Roofline: factored algorithm = 8.2 GFLOP; >=123 MB compulsory HBM traffic
// -> ~5.3us floor at 23.3 TB/s, so f32 WMMA (V_WMMA_F32_16X16X4_F32) keeps
// full reference precision while staying near the memory roof. Both GEMM
// stages are fused per (n, 8-timestep tile) through LDS so the 184 MB "xa"
// intermediate never touches HBM. Per-channel bias b.sum(0) cancels exactly
// under BN mean subtraction -> skipped. BN statistics accumulate in-kernel
// (LDS ds_add_f32 partials + 128 global atomics/block); the finalize pass's
// y round trip mostly stays in the 192 MB L2.
//
// LDS layout (all fragment accesses bank-conflict-free, b64/b128 vectorized):
//   Xs   [64][260]   row pad 4  -> A-frags: ds_load_2addr_b64, banks 4m+{0..3}
//   XA_t [256][68]   transposed -> B-frags adjacent-pair b64; D-store 2x b128
//   A_t  [3][32][36] transposed -> B-frags adjacent-pair b64
// ---------------------------------------------------------------------------

typedef __attribute__((ext_vector_type(2))) float v2f;
typedef __attribute__((ext_vector_type(4))) float v4f;
typedef __attribute__((ext_vector_type(8))) float v8f;

#define Nn 32
#define Cc 64
#define Tt 300
#define Vv 25
#define Oo 64
#define Kk 3
#define TT 8                       // timesteps per workgroup
#define NC (TT * 32)               // 256 padded (t,w) columns per tile
#define XP 260                     // Xs row pitch (256 + 4 pad)
#define AP 36                      // A_t row pitch (32 + 4 pad)
#define CP 68                      // XA_t row pitch (64 + 4 pad)
#define TB ((Tt + TT - 1) / TT)    // 38 time blocks

__global__ __launch_bounds__(256) void gcn_main_kernel(
    const float* __restrict__ x,     // [N,C,T,V]
    const float* __restrict__ Ag,    // [K,V,V]
    const float* __restrict__ Wg,    // [K,O,C]
    float* __restrict__ out,         // [N,O,T,V] (raw y before BN)
    float* __restrict__ stats)       // [2*O] sum / sumsq accumulators
{
    __shared__ float Xs[Cc * XP];        // input tile, row-padded   (65.0 KB)
    __shared__ float XAt[NC * CP];       // xa_k tile, TRANSPOSED    (68.0 KB)
    __shared__ float At[Kk * 32 * AP];   // A transposed + padded    (13.5 KB)
    __shared__ float sred[2 * Oo];       // per-block channel sum/sumsq

    const int tid  = threadIdx.x;
    const int wv   = tid >> 5;         // wave id 0..7
    const int lane = tid & 31;
    const int hi   = lane >> 4;        // lane-half select
    const int mn   = lane & 15;        // M (A-frag row) / N (B,C,D col)
    const int kb   = hi << 1;          // K base for A/B fragments (0 or 2)

    const int n_idx = blockIdx.x / TB;
    const int t0    = (blockIdx.x % TB) * TT;

    if (tid < 2 * Oo) sred[tid] = 0.0f;

    // ---- load X tile [C][TT*32] (row pitch XP), zero pad v>=25, t>=300 ----
    for (int i = tid; i < Cc * NC; i += 256) {
        int c   = i >> 8;          // NC == 256
        int col = i & (NC - 1);
        int tl  = col >> 5;
        int v   = col & 31;
        int t   = t0 + tl;
        float val = 0.0f;
        if (v < Vv && t < Tt)
            val = x[((n_idx * Cc + c) * Tt + t) * Vv + v];
        Xs[c * XP + col] = val;
    }
    // ---- load A transposed: At[k][w][v] = A[k][v][w], zero padded ----
    for (int i = tid; i < Kk * 32 * 32; i += 256) {
        int k = i >> 10;
        int w = (i >> 5) & 31;
        int v = i & 31;
        float val = 0.0f;
        if (w < Vv && v < Vv)
            val = Ag[(k * Vv + v) * Vv + w];
        At[(k * 32 + w) * AP + v] = val;
    }

    const v8f vzero = {};
    const int o0  = (wv >> 1) << 4;    // stage-2 output-channel strip
    const int cth = (wv & 1) << 3;     // stage-2 column-tile half

    v8f acc[8];
    #pragma unroll
    for (int j = 0; j < 8; ++j) acc[j] = vzero;

    for (int k = 0; k < Kk; ++k) {
        __syncthreads();   // XAt free for reuse / Xs+At visible

        // ---- stage 1: XA_k[(t,w), c] = sum_v X[c,(t,v)] * A_k[v,w] ----
        // 64 output tiles (4 c-strips x 16 col-tiles), 8 per wave.
        {
            #pragma unroll
            for (int i = 0; i < 8; ++i) {
                int ti = (wv << 3) + i;
                int c0 = (ti >> 4) << 4;      // c strip
                int ct = ti & 15;             // column tile
                int tl = ct >> 1;             // local timestep
                int w0 = (ct & 1) << 4;       // w half
                const float* xrow = &Xs[(c0 + mn) * XP + (tl << 5)];
                const float* arow = &At[(k * 32 + w0 + mn) * AP];
                v8f cc = vzero;
                #pragma unroll
                for (int v0 = 0; v0 < 32; v0 += 4) {
                    v2f af = *(const v2f*)&xrow[v0 + kb];
                    v2f bf = *(const v2f*)&arow[v0 + kb];
                    cc = __builtin_amdgcn_wmma_f32_16x16x4_f32(
                        false, af, false, bf, (short)0, cc, false, false);
                }
                // transposed store: 8 consecutive c's -> two ds_store_b128
                v4f lo = { cc[0], cc[1], cc[2], cc[3] };
                v4f hi4 = { cc[4], cc[5], cc[6], cc[7] };
                float* dst = &XAt[((ct << 4) + mn) * CP + c0 + (hi << 3)];
                *(v4f*)dst = lo;
                *((v4f*)dst + 1) = hi4;
            }
        }
        __syncthreads();   // XA_k complete

        // ---- stage 2: Y[o,(t,w)] += W_k[o,c] @ XA_k[c,(t,w)] ----
        // Each wave: fixed 16-row o strip, 8 column tiles, K=64 reduction.
        {
            const float* Wk = &Wg[((k * Oo) + o0 + mn) * Cc];
            v2f wf[16];
            #pragma unroll
            for (int i = 0; i < 16; ++i)
                wf[i] = *(const v2f*)&Wk[(i << 2) + kb];
            #pragma unroll
            for (int tj = 0; tj < 8; ++tj) {
                const float* xcol = &XAt[(((cth + tj) << 4) + mn) * CP];
                v8f a = acc[tj];
                #pragma unroll
                for (int c4 = 0; c4 < 16; ++c4) {
                    v2f bf = *(const v2f*)&xcol[(c4 << 2) + kb];
                    a = __builtin_amdgcn_wmma_f32_16x16x4_f32(
                        false, wf[c4], false, bf, (short)0, a, false, false);
                }
                acc[tj] = a;
            }
        }
    }

    // ---- store raw y + accumulate per-channel BN partial stats ----
    {
        float s[8], q[8];
        #pragma unroll
        for (int j = 0; j < 8; ++j) { s[j] = 0.0f; q[j] = 0.0f; }

        #pragma unroll
        for (int tj = 0; tj < 8; ++tj) {
            int col = ((cth + tj) << 4) + mn;
            int tl  = col >> 5;
            int vv  = col & 31;
            int t   = t0 + tl;
            if (vv < Vv && t < Tt) {
                #pragma unroll
                for (int j = 0; j < 8; ++j) {
                    int o = o0 + j + (hi << 3);
                    float val = acc[tj][j];
                    out[((n_idx * Oo + o) * Tt + t) * Vv + vv] = val;
                    s[j] += val;
                    q[j] += val * val;
                }
            }
        }
        #pragma unroll
        for (int j = 0; j < 8; ++j) {
            int o = o0 + j + (hi << 3);
            atomicAdd(&sred[o], s[j]);          // ds_add_f32
            atomicAdd(&sred[Oo + o], q[j]);
        }
    }
    __syncthreads();
    if (tid < Oo) {
        atomicAdd(&stats[tid], sred[tid]);
        atomicAdd(&stats[Oo + tid], sred[Oo + tid]);
    }
}

__global__ void zero_stats_kernel(float* __restrict__ stats) {
    if (threadIdx.x < 2 * Oo) stats[threadIdx.x] = 0.0f;
}

// One block per (n,o) segment: 7500 contiguous floats = 1875 float4's,
// uniform scale/shift -> pure b128 streaming through L2.
__global__ __launch_bounds__(256) void bn_finalize_kernel(
    float* __restrict__ out, const float* __restrict__ stats,
    const float* __restrict__ gamma, const float* __restrict__ beta)
{
    const int seg = blockIdx.x;          // n*Oo + o
    const int o   = seg & (Oo - 1);
    const float Mcnt = (float)(Nn * Tt * Vv);   // 240000 samples per channel
    float mean = stats[o] / Mcnt;
    float var  = stats[Oo + o] / Mcnt - mean * mean;
    float inv  = rsqrtf(var + 1e-5f);
    float scv  = gamma[o] * inv;
    float shv  = beta[o] - mean * scv;

    float4* p = (float4*)(out + (size_t)seg * (Tt * Vv));
    const int nvec = (Tt * Vv) / 4;      // 1875
    for (int i = threadIdx.x; i < nvec; i += 256) {
        float4 v = p[i];
        v.x = fmaxf(v.x * scv + shv, 0.0f);
        v.y = fmaxf(v.y * scv + shv, 0.0f);
        v.z = fmaxf(v.z * scv + shv, 0.0f);
        v.w = fmaxf(v.w * scv + shv, 0.0f);
        p[i] = v;
    }
}

extern "C" void kernel_launch(void* const* d_in, const int* in_sizes, int n_in,
                              void* d_out, int out_size, void* d_ws, size_t ws_size,
                              hipStream_t stream) {
    const float* x     = (const float*)d_in[0];
    const float* A     = (const float*)d_in[1];
    const float* W     = (const float*)d_in[2];
    // d_in[3] = b : per-channel constant shift, cancelled exactly by BN mean.
    const float* gamma = (const float*)d_in[4];
    const float* beta  = (const float*)d_in[5];
    float* out   = (float*)d_out;
    float* stats = (float*)d_ws;     // 128 floats of scratch

    zero_stats_kernel<<<dim3(1), dim3(128), 0, stream>>>(stats);
    gcn_main_kernel<<<dim3(Nn * TB), dim3(256), 0, stream>>>(x, A, W, out, stats);
    bn_finalize_kernel<<<dim3(Nn * Oo), dim3(256), 0, stream>>>(out, stats, gamma, beta);
}